// NeuralIncidenceMask_45921790329371
// MI455X (gfx1250) — compile-verified
//
#include <hip/hip_runtime.h>
#include <math.h>

#define RANK 16
#define FEAT 256

typedef __attribute__((ext_vector_type(2))) float v2f;
typedef __attribute__((ext_vector_type(8))) float v8f;

// Order-preserving float -> u32 key (larger float => larger key)
__device__ __forceinline__ unsigned key_of(float f) {
  unsigned u = __float_as_uint(f);
  return (u & 0x80000000u) ? ~u : (u | 0x80000000u);
}

// ---------------------------------------------------------------------------
// proj[N,16] = x[N,256] @ W[16,256]^T  via V_WMMA_F32_16X16X4_F32.
// One wave per 16-row tile; 64 WMMA steps over K=256 (K=4 per step).
// A layout (16x4 f32): lanes 0-15 hold K=0(v0),K=1(v1); lanes 16-31 hold K=2,K=3.
// B layout (4x16): lane n holds column n; B[k][n] = W[n][k] -> same load pattern.
// D layout (16x16 f32): VGPR j -> M=j (lanes 0-15), M=j+8 (lanes 16-31), N=lane&15.
// ---------------------------------------------------------------------------
__global__ void proj_wmma_kernel(const float* __restrict__ x,
                                 const float* __restrict__ W,
                                 float* __restrict__ proj,
                                 int num_tiles) {
  const int wave = threadIdx.x >> 5;                 // 8 waves / 256-thread block
  const int tile = blockIdx.x * 8 + wave;
  if (tile >= num_tiles) return;                     // wave-uniform guard
  const int lane = threadIdx.x & 31;
  const int half = lane >> 4;                        // 0: lanes 0-15, 1: lanes 16-31
  const int m    = lane & 15;

  const float* __restrict__ xrow = x + (size_t)(tile * 16 + m) * FEAT;
  const float* __restrict__ wrow = W + (size_t)m * FEAT;

  v8f acc = {};
#pragma unroll 4
  for (int k = 0; k < FEAT; k += 4) {
    v2f a = *(const v2f*)(xrow + k + 2 * half);      // A[m][k+2h], A[m][k+2h+1]
    v2f b = *(const v2f*)(wrow + k + 2 * half);      // B[k+2h][m], B[k+2h+1][m]
    acc = __builtin_amdgcn_wmma_f32_16x16x4_f32(
        /*neg_a=*/false, a, /*neg_b=*/false, b,
        /*c_mod=*/(short)0, acc, /*reuse_a=*/false, /*reuse_b=*/false);
  }

  float* out = proj + (size_t)tile * 16 * RANK;
#pragma unroll
  for (int j = 0; j < 8; ++j)
    out[(size_t)(j + 8 * half) * RANK + m] = acc[j];
}

// ---------------------------------------------------------------------------
// Workspace init: zero per-edge accumulators + histogram, seed select state.
// ---------------------------------------------------------------------------
__global__ void init_ws_kernel(float* sum_scores, float* sum_hard, float* cnt,
                               unsigned* hist, unsigned* state,
                               const int* __restrict__ kptr, int num_he) {
  int i = blockIdx.x * blockDim.x + threadIdx.x;
  if (i < num_he) { sum_scores[i] = 0.f; sum_hard[i] = 0.f; cnt[i] = 0.f; }
  if (i < 256) hist[i] = 0u;
  if (i == 0) {
    state[0] = 0u;                      // radix prefix
    state[1] = (unsigned)kptr[0];       // remaining k
    state[2] = 0u;                      // tie-claim counter
  }
}

// ---------------------------------------------------------------------------
// Per-incidence: gather proj row + edge factor row (L2-resident tables),
// rank-16 dot, sigmoid, accumulate per-edge count / score sum.
// ---------------------------------------------------------------------------
__global__ void score_kernel(const float* __restrict__ proj,
                             const float* __restrict__ ef,
                             const int* __restrict__ V_idx,
                             const int* __restrict__ E_idx,
                             const int* __restrict__ edge_ids,
                             float* __restrict__ scores,
                             float* sum_scores, float* cnt, int nnz) {
  int i = blockIdx.x * blockDim.x + threadIdx.x;
  if (i >= nnz) return;
  const int v  = V_idx[i];
  const int e  = E_idx[i];
  const int he = edge_ids[e];
  const float4* p = (const float4*)(proj + (size_t)v * RANK);
  const float4* f = (const float4*)(ef + (size_t)he * RANK);
  float acc = 0.f;
#pragma unroll
  for (int j = 0; j < 4; ++j) {
    float4 a = p[j], b = f[j];
    acc += a.x * b.x + a.y * b.y + a.z * b.z + a.w * b.w;
  }
  float s = 1.0f / (1.0f + expf(-acc));
  scores[i] = s;
  atomicAdd(&sum_scores[e], s);
  atomicAdd(&cnt[e], 1.0f);
}

// ---------------------------------------------------------------------------
// Radix-select pass `pass` (byte 3..0): histogram current byte among elements
// whose higher key bytes match the running prefix. LDS-privatized histogram.
// ---------------------------------------------------------------------------
__global__ void hist_kernel(const float* __restrict__ scores, unsigned* hist,
                            const unsigned* __restrict__ state, int pass, int nnz) {
  __shared__ unsigned sh[256];
  sh[threadIdx.x] = 0u;
  __syncthreads();
  const unsigned prefix = state[0];
  const int shift = 24 - 8 * pass;
  int i = blockIdx.x * blockDim.x + threadIdx.x;
  if (i < nnz) {
    unsigned key = key_of(scores[i]);
    unsigned long long diff = (unsigned long long)(key ^ prefix);
    if ((diff >> (shift + 8)) == 0ULL)
      atomicAdd(&sh[(key >> shift) & 255u], 1u);
  }
  __syncthreads();
  unsigned c = sh[threadIdx.x];
  if (c) atomicAdd(&hist[threadIdx.x], c);
}

// Single block: walk bins high->low, pick bin holding the remaining-k'th
// largest, fold into prefix, then zero histogram for the next pass.
__global__ void select_kernel(unsigned* hist, unsigned* state, int pass) {
  __shared__ unsigned sh[256];
  sh[threadIdx.x] = hist[threadIdx.x];
  __syncthreads();
  if (threadIdx.x == 0) {
    unsigned rem = state[1];
    const int shift = 24 - 8 * pass;
    unsigned cum = 0; int sel = 0;
    for (int b = 255; b >= 0; --b) {
      unsigned c = sh[b];
      if (cum + c >= rem) { sel = b; rem -= cum; break; }
      cum += c;
    }
    state[0] |= ((unsigned)sel << shift);
    state[1] = rem;   // after pass 3: #ties at threshold that must be kept
  }
  hist[threadIdx.x] = 0u;
}

// ---------------------------------------------------------------------------
// Mark keep set: key > T -> 1; key == T -> claim one of `rem` tie slots.
// soft == hard in the forward pass (scores - stop_grad(scores) == 0).
// ---------------------------------------------------------------------------
__global__ void mark_kernel(const float* __restrict__ scores,
                            const int* __restrict__ E_idx,
                            unsigned* state,
                            float* __restrict__ soft, float* __restrict__ hard,
                            float* sum_hard, int nnz) {
  int i = blockIdx.x * blockDim.x + threadIdx.x;
  if (i >= nnz) return;
  const unsigned T = state[0];
  const unsigned rem = state[1];
  const unsigned key = key_of(scores[i]);
  float h = 0.f;
  if (key > T) {
    h = 1.f;
  } else if (key == T) {
    unsigned slot = atomicAdd(&state[2], 1u);
    h = (slot < rem) ? 1.f : 0.f;
  }
  hard[i] = h;
  soft[i] = h;
  if (h != 0.f) atomicAdd(&sum_hard[E_idx[i]], 1.0f);
}

__global__ void finalize_kernel(const float* __restrict__ sum_scores,
                                const float* __restrict__ sum_hard,
                                const float* __restrict__ cnt,
                                float* __restrict__ edge_probs,
                                float* __restrict__ edge_soft,
                                float* __restrict__ edge_hard, int num_he) {
  int e = blockIdx.x * blockDim.x + threadIdx.x;
  if (e >= num_he) return;
  float d = fmaxf(cnt[e], 1.0f);
  edge_probs[e] = sum_scores[e] / d;
  float m = sum_hard[e] / d;
  edge_soft[e] = m;
  edge_hard[e] = (m > 0.f) ? 1.f : 0.f;
}

// ---------------------------------------------------------------------------
extern "C" void kernel_launch(void* const* d_in, const int* in_sizes, int n_in,
                              void* d_out, int out_size, void* d_ws, size_t ws_size,
                              hipStream_t stream) {
  const float* x        = (const float*)d_in[0];
  const float* W        = (const float*)d_in[1];
  const float* ef       = (const float*)d_in[2];
  const int*   V_idx    = (const int*)d_in[3];
  const int*   E_idx    = (const int*)d_in[4];
  const int*   edge_ids = (const int*)d_in[5];
  const int*   kptr     = (const int*)d_in[6];

  const int num_nodes = in_sizes[0] / FEAT;     // 100000
  const int nnz       = in_sizes[3];            // 4000000
  const int num_he    = in_sizes[5];            // 125000

  float* out        = (float*)d_out;
  float* scores     = out;
  float* soft       = out + (size_t)nnz;
  float* hard       = out + (size_t)2 * nnz;
  float* edge_probs = out + (size_t)3 * nnz;
  float* edge_soft  = edge_probs + num_he;
  float* edge_hard  = edge_soft + num_he;

  // Workspace layout (~7.9 MB)
  float*    proj       = (float*)d_ws;
  float*    sum_scores = proj + (size_t)num_nodes * RANK;
  float*    sum_hard   = sum_scores + num_he;
  float*    cnt        = sum_hard + num_he;
  unsigned* hist       = (unsigned*)(cnt + num_he);
  unsigned* state      = hist + 256;

  const int heBlocks = (num_he + 255) / 256;
  init_ws_kernel<<<heBlocks, 256, 0, stream>>>(sum_scores, sum_hard, cnt,
                                               hist, state, kptr, num_he);

  const int num_tiles = (num_nodes + 15) / 16;  // 6250 (exact)
  proj_wmma_kernel<<<(num_tiles + 7) / 8, 256, 0, stream>>>(x, W, proj, num_tiles);

  const int nb = (nnz + 255) / 256;
  score_kernel<<<nb, 256, 0, stream>>>(proj, ef, V_idx, E_idx, edge_ids,
                                       scores, sum_scores, cnt, nnz);

  for (int p = 0; p < 4; ++p) {
    hist_kernel<<<nb, 256, 0, stream>>>(scores, hist, state, p, nnz);
    select_kernel<<<1, 256, 0, stream>>>(hist, state, p);
  }

  mark_kernel<<<nb, 256, 0, stream>>>(scores, E_idx, state, soft, hard,
                                      sum_hard, nnz);
  finalize_kernel<<<heBlocks, 256, 0, stream>>>(sum_scores, sum_hard, cnt,
                                                edge_probs, edge_soft,
                                                edge_hard, num_he);
}